// DistributedMemory_30545807409973
// MI455X (gfx1250) — compile-verified
//
#include <hip/hip_runtime.h>

// CDNA5 / gfx1250 — wave32, WMMA 16x16x4 f32 used as a 16-wide dot-product
// reduction engine (A broadcast over M).

typedef __attribute__((ext_vector_type(2))) float v2f;
typedef __attribute__((ext_vector_type(8))) float v8f;

#define VEC_DIM 128
#define N_WORDS_C 100000
#define CTX 8
#define NS 16
#define WAVES_PER_BLOCK 8
#define THREADS_PER_BLOCK (WAVES_PER_BLOCK * 32)

__global__ __launch_bounds__(THREADS_PER_BLOCK) void
dm_doc2vec_kernel(const int* __restrict__ doc_ids,
                  const int* __restrict__ context_ids,
                  const int* __restrict__ sample_ids,
                  const float* __restrict__ para,     // [N_DOCS, 128]
                  const float* __restrict__ word,     // [N_WORDS, 128]
                  const float* __restrict__ outs,     // [128, N_WORDS]
                  float* __restrict__ out,            // [B, NS]
                  int Btot)
{
    __shared__ float lds_x[WAVES_PER_BLOCK * VEC_DIM];  // 4 KB

    const int lane = threadIdx.x & 31;
    const int wave = threadIdx.x >> 5;
    const int b    = blockIdx.x * WAVES_PER_BLOCK + wave;
    const bool active = (b < Btot);   // wave-uniform

    // ---- Phase 1: x[b,:] = para[doc] + sum_c word[ctx[c]] ; lane owns 4 dims.
    if (active) {
        const float* prow = para + (size_t)doc_ids[b] * VEC_DIM;
        float4 xa = *(const float4*)(prow + 4 * lane);
        #pragma unroll
        for (int c = 0; c < CTX; ++c) {
            const float* wrow = word + (size_t)context_ids[b * CTX + c] * VEC_DIM;
            float4 wv = *(const float4*)(wrow + 4 * lane);
            xa.x += wv.x; xa.y += wv.y; xa.z += wv.z; xa.w += wv.w;
        }
        *(float4*)(&lds_x[wave * VEC_DIM + 4 * lane]) = xa;
    }
    __syncthreads();
    if (!active) return;  // whole wave exits together -> EXEC all-1 for WMMA below

    // ---- Phase 2: y[s] = sum_d x[d] * outs[d, w_s] via 32 chained WMMAs.
    // Lane roles: s = lane&15 selects sampled column, half = lane>>4 selects
    // which K-pair of the 16x16x4 this lane supplies.
    const int s    = lane & 15;
    const int half = lane >> 4;
    const int w    = sample_ids[b * NS + s];
    const float* pcol  = outs + w;                    // column w, stride N_WORDS
    const float* xbase = &lds_x[wave * VEC_DIM];

    v8f acc = {};
    #pragma unroll
    for (int t = 0; t < VEC_DIM / 4; ++t) {
        const int dk = 4 * t + 2 * half;              // K rows this lane feeds
        // A (broadcast over M): VGPR0 = x[dk], VGPR1 = x[dk+1]  (ds_load_b64 broadcast)
        v2f a = *(const v2f*)(xbase + dk);
        // B: VGPR0 = O[dk, w], VGPR1 = O[dk+1, w]
        v2f bm;
        bm.x = pcol[(size_t)dk * N_WORDS_C];
        bm.y = pcol[(size_t)(dk + 1) * N_WORDS_C];
        acc = __builtin_amdgcn_wmma_f32_16x16x4_f32(
            /*neg_a=*/false, a, /*neg_b=*/false, bm,
            /*c_mod=*/(short)0, acc, /*reuse_a=*/false, /*reuse_b=*/false);
    }

    // D[m, n]: every row m equals y[:]; D-VGPR0 holds M=0 row for lanes 0..15.
    if (lane < NS) out[b * NS + lane] = acc[0];
}

extern "C" void kernel_launch(void* const* d_in, const int* in_sizes, int n_in,
                              void* d_out, int out_size, void* d_ws, size_t ws_size,
                              hipStream_t stream) {
    const int*   doc_ids     = (const int*)d_in[0];
    const int*   context_ids = (const int*)d_in[1];
    const int*   sample_ids  = (const int*)d_in[2];
    const float* para        = (const float*)d_in[3];
    const float* word        = (const float*)d_in[4];
    const float* outs        = (const float*)d_in[5];
    float*       out         = (float*)d_out;

    const int Btot   = in_sizes[0];                       // 16384
    const int blocks = (Btot + WAVES_PER_BLOCK - 1) / WAVES_PER_BLOCK;

    dm_doc2vec_kernel<<<blocks, THREADS_PER_BLOCK, 0, stream>>>(
        doc_ids, context_ids, sample_ids, para, word, outs, out, Btot);
}